// TransformedInput_15221364097579
// MI455X (gfx1250) — compile-verified
//
#include <hip/hip_runtime.h>

// Zonotope construction for x:[8,3,32,32] -> z:[8, 1+3*32*32, 3, 32, 32]
//   plane 0                 : center(x)
//   plane 1+hw, pos (c,h,w) : err(x[b,c,h,w])   (spatial-only diagonal)
//   everything else         : 0
//
// Memory-bound: ~302 MB of stores vs ~96 KB of loads -> ~13 us floor at
// 23.3 TB/s. K1 streams zeros with non-temporal b128 stores (output > 192 MB
// L2 and never re-read); K2 scatters the 49,152 nonzero values.

typedef __attribute__((ext_vector_type(4))) float v4f;

#define NB 8
#define NC 3
#define NH 32
#define NW 32
#define HW (NH * NW)                      // 1024
#define NPLANE (1 + NC * HW)              // 3073
#define CHW (NC * HW)                     // 3072 floats per plane
#define BSTRIDE ((long long)NPLANE * CHW) // 9,440,256 floats per batch image

// Each thread fills 64 consecutive bytes (4 x float4) per iteration so the
// compiler emits 4 back-to-back global_store_b128 th:TH_STORE_NT with
// immediate offsets 0/16/32/48 off one address pair; a wave32 covers a
// contiguous 2 KB per iteration. n16 = number of 16-float chunks.
__global__ void zono_fill_zero(float* __restrict__ out, unsigned n16) {
  unsigned i = blockIdx.x * blockDim.x + threadIdx.x;
  const unsigned stride = gridDim.x * blockDim.x;
  v4f z = {0.0f, 0.0f, 0.0f, 0.0f};
  for (; i < n16; i += stride) {
    v4f* p = (v4f*)(out + (unsigned long long)i * 16u);
    __builtin_nontemporal_store(z, p + 0);
    __builtin_nontemporal_store(z, p + 1);
    __builtin_nontemporal_store(z, p + 2);
    __builtin_nontemporal_store(z, p + 3);
  }
}

__global__ void zono_scatter(const float* __restrict__ x, float* __restrict__ out) {
  int t = blockIdx.x * blockDim.x + threadIdx.x; // 0 .. NB*NC*HW-1 (grid sized exactly)
  float v = x[t];                                // x is [B,C,H,W] contiguous -> flat == t

  const float eps = 0.1f;
  const float one = 1.0f;
  bool lo = v < eps;
  bool hi = v > (one - eps);

  // Match reference op order exactly:
  //   lo: center=(x+eps)*0.5      err=(eps+x)*0.5
  //   hi: center=((x+1)-eps)*0.5  err=((1-x)+eps)*0.5
  //   else: center=x              err=eps
  float center = lo ? (v + eps) * 0.5f
                    : (hi ? ((v + one) - eps) * 0.5f : v);
  float err    = lo ? (eps + v) * 0.5f
                    : (hi ? ((one - v) + eps) * 0.5f : eps);

  int b  = t / CHW;
  int r  = t - b * CHW;
  int c  = r >> 10;        // r / HW   (HW = 1024)
  int hw = r & (HW - 1);   // r % HW

  long long base = (long long)b * BSTRIDE;
  // plane 0: center at (c, h, w)
  out[base + (long long)c * HW + hw] = center;
  // plane 1+hw: err at (c, h, w)
  out[base + (long long)(1 + hw) * CHW + (long long)c * HW + hw] = err;
}

extern "C" void kernel_launch(void* const* d_in, const int* in_sizes, int n_in,
                              void* d_out, int out_size, void* d_ws, size_t ws_size,
                              hipStream_t stream) {
  const float* x = (const float*)d_in[0];
  float* out = (float*)d_out;

  // out_size = 8 * 3073 * 3072 = 75,522,048 floats; divisible by 16.
  unsigned n16 = (unsigned)(out_size >> 4); // 4,720,128 chunks of 64 B

  // 4096 blocks x 256 threads (8 wave32/block) -> ~1M threads, ~4-5 chunk
  // iterations each (16-20 b128 NT stores); saturates HBM across all WGPs
  // while amortizing loop overhead 4x per store group.
  zono_fill_zero<<<4096, 256, 0, stream>>>(out, n16);

  const int n = NB * NC * HW; // 24576
  zono_scatter<<<n / 256, 256, 0, stream>>>(x, out);
}